// SE_RGCN_15367392985225
// MI455X (gfx1250) — compile-verified
//
#include <hip/hip_runtime.h>

#define N_NODES 50000
#define R_REL   64
#define E_EDGE  20000
#define B_BASES 16
#define DIM_IN  256
#define DIM_OUT 256
#define EPB     32                    // edges per chunk (two 16-row M tiles)
#define CPB     5                     // chunks per block (625 = 125 * 5)
#define GRIDX   (E_EDGE / EPB / CPB)  // 125

typedef float v2f __attribute__((ext_vector_type(2)));
typedef float v8f __attribute__((ext_vector_type(8)));

// LDS row stride for A tiles: 260 floats -> bank(m,k) = (4m+k)%64 distinct for
// m=0..15 within a half-wave column read; 1040B rows stay float4-aligned.
#define LDS_STRIDE 260

#define AS3 __attribute__((address_space(3)))

// ---------------------------------------------------------------------------
// Async global->LDS copy of 64 floats (one thread's share of a gathered row)
// via CDNA5 GLOBAL_LOAD_ASYNC_TO_LDS_B128 (ASYNCcnt-tracked).  INST_OFFSET is
// added to BOTH the LDS and global addresses (ISA 08, §4.4), so one base pair
// + immediate offsets covers the whole 256B segment.  Inline asm is used
// because the clang builtin's signature differs across toolchains.
// ---------------------------------------------------------------------------
__device__ __forceinline__ void gather_seg(const float* g, float* l) {
  const unsigned lofs = (unsigned)(uintptr_t)(AS3 float*)l;  // LDS byte addr
  #pragma unroll
  for (int i = 0; i < 16; ++i) {
    asm volatile("global_load_async_to_lds_b128 %0, %1, off offset:%2"
                 :
                 : "v"(lofs), "v"(g), "i"(i * 16)
                 : "memory");
  }
}

__device__ __forceinline__ void gather_drain() {
  asm volatile("s_wait_asynccnt 0x0" ::: "memory");
}

// ---------------------------------------------------------------------------
// Kernel 1: basis decomposition, stored TRANSPOSED:
//   wT[r][o][i] = sum_b att[r,b] * basis[b][i][o]
// ---------------------------------------------------------------------------
__global__ __launch_bounds__(256)
void rgcn_basis_T(const float* __restrict__ att, const float* __restrict__ basis,
                  float* __restrict__ wT) {
  const int i = threadIdx.x;        // input-dim index (contiguous in wT row)
  const int o = blockIdx.x;         // output-dim index
  const int r = blockIdx.y;
  float acc = 0.f;
  #pragma unroll
  for (int b = 0; b < B_BASES; ++b)
    acc = fmaf(att[r * B_BASES + b],
               basis[(size_t)b * (DIM_IN * DIM_OUT) + (size_t)i * DIM_OUT + o],
               acc);
  wT[(size_t)r * (DIM_IN * DIM_OUT) + (size_t)o * DIM_IN + i] = acc;
}

// ---------------------------------------------------------------------------
// Kernel 1b: rootT[o][i] = root[i][o]  (16x16 LDS tile transpose)
// ---------------------------------------------------------------------------
__global__ __launch_bounds__(256)
void rgcn_transpose(const float* __restrict__ src, float* __restrict__ dst) {
  __shared__ float t[16][17];
  const int lx = threadIdx.x & 15;
  const int ly = threadIdx.x >> 4;
  const int bo = blockIdx.x * 16;
  const int bi = blockIdx.y * 16;
  t[ly][lx] = src[(size_t)(bi + ly) * DIM_OUT + bo + lx];
  __syncthreads();
  dst[(size_t)(bo + ly) * DIM_IN + bi + lx] = t[lx][ly];
}

// ---------------------------------------------------------------------------
// Wave-level GEMM: MT stacked 16x16 fp32 tiles over K=256, one shared B
// fragment per K-step.  BT is transposed [N][K] so a lane's (kk,kk+1) B
// fragment is one aligned global_load_b64.
// ---------------------------------------------------------------------------
template <int MT>
__device__ __forceinline__ void wave_tile_gemm(const float* lds_a,
                                               const float* __restrict__ BT,
                                               int n0, int lane, v8f acc[MT]) {
  const int m     = lane & 15;
  const int khalf = (lane >> 4) << 1;   // lanes 16..31 hold K+2,K+3
  const int n     = n0 + m;
  const float* __restrict__ brow = BT + (size_t)n * DIM_IN;
  #pragma unroll 8
  for (int k0 = 0; k0 < DIM_IN; k0 += 4) {
    const int kk = k0 + khalf;
    const v2f b = *(const v2f*)(brow + kk);          // global_load_b64
    #pragma unroll
    for (int mt = 0; mt < MT; ++mt) {
      v2f a;
      a.x = lds_a[(mt * 16 + m) * LDS_STRIDE + kk];  // ds_load_2addr_b64
      a.y = lds_a[(mt * 16 + m) * LDS_STRIDE + kk + 1];
      acc[mt] = __builtin_amdgcn_wmma_f32_16x16x4_f32(false, a, false, b,
                                                      (short)0, acc[mt],
                                                      false, false);
    }
  }
}

// ---------------------------------------------------------------------------
// Kernel 2: out = x @ root  (initializes every element of d_out)
// ---------------------------------------------------------------------------
__global__ __launch_bounds__(128)
void rgcn_root_gemm(const float* __restrict__ x, const float* __restrict__ rootT,
                    float* __restrict__ out) {
  __shared__ float xa[16 * LDS_STRIDE];
  const int tid    = threadIdx.x;
  const int m_base = blockIdx.x * 16;

  {
    const int row = tid >> 3;
    const int c0  = (tid & 7) * 32;
    const float4* g = (const float4*)(x + (size_t)(m_base + row) * DIM_IN + c0);
    float4* l = (float4*)(xa + row * LDS_STRIDE + c0);
    #pragma unroll
    for (int i = 0; i < 8; ++i) l[i] = g[i];
  }
  __syncthreads();

  const int lane   = tid & 31;
  const int wave   = tid >> 5;
  const int rowoff = (lane >> 4) << 3;
  #pragma unroll
  for (int nt = 0; nt < 4; ++nt) {
    const int n0 = wave * 64 + nt * 16;
    v8f acc[1] = {{0.f, 0.f, 0.f, 0.f, 0.f, 0.f, 0.f, 0.f}};
    wave_tile_gemm<1>(xa, rootT, n0, lane, acc);
    const int n = n0 + (lane & 15);
    #pragma unroll
    for (int v = 0; v < 8; ++v)
      out[(size_t)(m_base + v + rowoff) * DIM_OUT + n] = acc[0][v];
  }
}

// ---------------------------------------------------------------------------
// Kernel 3 (persistent, double-buffered): each block owns CPB consecutive
// 32-edge chunks of one relation.  The async gather for chunk j+1 streams
// into the alternate LDS buffer while chunk j's WMMA + atomic scatter runs.
// ---------------------------------------------------------------------------
__global__ __launch_bounds__(128)
void rgcn_edge_gemm(const float* __restrict__ x, const float* __restrict__ wT,
                    const int* __restrict__ esrc, const int* __restrict__ edst,
                    const float* __restrict__ deg, float* __restrict__ out) {
  __shared__ float xa[2][EPB * LDS_STRIDE];
  __shared__ int   ssrc[2][EPB];
  __shared__ int   sdst[2][EPB];
  __shared__ float sinv[2][EPB];

  const int tid    = threadIdx.x;
  const int r      = blockIdx.y;
  const int c_base = blockIdx.x * CPB;

  const float* __restrict__ BrT = wT + (size_t)r * (DIM_IN * DIM_OUT);
  __builtin_prefetch(BrT + (size_t)tid * 512, 0, 3);   // global_prefetch_b8

  const int lane   = tid & 31;
  const int wave   = tid >> 5;
  const int rowoff = (lane >> 4) << 3;
  const int grow   = tid >> 2;          // gather row 0..31
  const int gcol   = (tid & 3) * 64;    // gather column start

  // meta for chunk 0 -> buf 0
  if (tid < EPB) {
    const size_t eidx = (size_t)r * E_EDGE + (size_t)c_base * EPB + tid;
    const int d = edst[eidx];
    ssrc[0][tid] = esrc[eidx];
    sdst[0][tid] = d;
    sinv[0][tid] = 1.0f / deg[d];
  }
  __syncthreads();
  // gather chunk 0 -> buf 0
  gather_seg(x + (size_t)ssrc[0][grow] * DIM_IN + gcol,
             &xa[0][grow * LDS_STRIDE + gcol]);

  for (int j = 0; j < CPB; ++j) {
    const int cur = j & 1;
    const int nxt = cur ^ 1;

    if (j + 1 < CPB && tid < EPB) {     // meta for next chunk
      const size_t eidx =
          (size_t)r * E_EDGE + (size_t)(c_base + j + 1) * EPB + tid;
      const int d = edst[eidx];
      ssrc[nxt][tid] = esrc[eidx];
      sdst[nxt][tid] = d;
      sinv[nxt][tid] = 1.0f / deg[d];
    }

    gather_drain();                     // s_wait_asynccnt 0
    __syncthreads();                    // cur tile ready; nxt meta visible

    if (j + 1 < CPB)                    // overlaps with compute below
      gather_seg(x + (size_t)ssrc[nxt][grow] * DIM_IN + gcol,
                 &xa[nxt][grow * LDS_STRIDE + gcol]);

    #pragma unroll
    for (int nt = 0; nt < 4; ++nt) {
      const int n0 = wave * 64 + nt * 16;
      v8f acc[2] = {{0.f, 0.f, 0.f, 0.f, 0.f, 0.f, 0.f, 0.f},
                    {0.f, 0.f, 0.f, 0.f, 0.f, 0.f, 0.f, 0.f}};
      wave_tile_gemm<2>(&xa[cur][0], BrT, n0, lane, acc);
      const int n = n0 + (lane & 15);
      #pragma unroll
      for (int mt = 0; mt < 2; ++mt) {
        #pragma unroll
        for (int v = 0; v < 8; ++v) {
          const int row = mt * 16 + v + rowoff;
          atomicAdd(&out[(size_t)sdst[cur][row] * DIM_OUT + n],
                    acc[mt][v] * sinv[cur][row]);
        }
      }
    }
    __syncthreads();   // all waves done with cur meta/tile before reuse
  }
}

// ---------------------------------------------------------------------------
extern "C" void kernel_launch(void* const* d_in, const int* in_sizes, int n_in,
                              void* d_out, int out_size, void* d_ws, size_t ws_size,
                              hipStream_t stream) {
  const float* x     = (const float*)d_in[0];   // [N, IN]
  const float* att   = (const float*)d_in[1];   // [R, B]
  const float* basis = (const float*)d_in[2];   // [B, IN, OUT]
  const float* root  = (const float*)d_in[3];   // [IN, OUT]
  const float* deg   = (const float*)d_in[4];   // [N, 1]
  const int*   esrc  = (const int*)d_in[5];     // [R, E]
  const int*   edst  = (const int*)d_in[6];     // [R, E]
  float* out   = (float*)d_out;                                  // [N, OUT]
  float* wT    = (float*)d_ws;                                   // 16 MB
  float* rootT = wT + (size_t)R_REL * DIM_IN * DIM_OUT;          // +256 KB

  // 1) basis decomposition -> wT (transposed weights in scratch)
  dim3 gw(DIM_OUT, R_REL);
  rgcn_basis_T<<<gw, 256, 0, stream>>>(att, basis, wT);

  // 1b) rootT = root^T
  dim3 gt(DIM_OUT / 16, DIM_IN / 16);
  rgcn_transpose<<<gt, 256, 0, stream>>>(root, rootT);

  // 2) out = x @ root (writes every output element; no pre-zero needed)
  rgcn_root_gemm<<<N_NODES / 16, 128, 0, stream>>>(x, rootT, out);

  // 3) edge messages + scaled atomic scatter (stream-ordered after 1,1b,2)
  dim3 ge(GRIDX, R_REL);
  rgcn_edge_gemm<<<ge, 128, 0, stream>>>(x, wT, esrc, edst, deg, out);
}